// Memory_sup_23570780520960
// MI455X (gfx1250) — compile-verified
//
#include <hip/hip_runtime.h>
#include <math.h>

typedef __attribute__((ext_vector_type(2))) float v2f;
typedef __attribute__((ext_vector_type(8))) float v8f;

#define Bsz 4
#define Dch 256
#define Mcl 19
#define Hq 96
#define Wqq 96
#define HWq (Hq*Wqq)          // 9216
#define Ntot (Bsz*HWq)        // 36864
#define CHW (Dch*HWq)         // 2359296
#define Hm 768
#define TEMP_INV 10.0f
#define BN_EPS 1e-5f

// workspace layout (float offsets)
#define WS_SCORE 0            // 19*36864 column-major score [m][n]
#define WS_INV   700416       // 36864 per-pixel 1/||q||
#define WS_CMAX  737280       // 19 (pad 32)
#define WS_CSUM  737312       // 19 (pad 32)
#define WS_P2    737344       // 20*256  (mem @ Wm^T, row 19 zero pad)
#define WS_NOM   742464       // 20*256
#define WS_DEN   747584       // 20 (pad 32)
#define WS_RL    747616       // [0]=nll sum, [1]=valid count

// output layout (float offsets)
#define OUT_UQ 0
#define OUT_SQ 9437184
#define OUT_SM 10137600
#define OUT_LS 10838016

#define QSTR 260
#define BSTR 260
#define WSTR 257
#define SMSTR 24
#define PSTR 260

// ---------------- init: zero accumulators ----------------
__global__ void k_init(float* __restrict__ ws) {
  int i = blockIdx.x * 256 + threadIdx.x;
  if (i < (WS_RL + 2 - WS_NOM)) ws[WS_NOM + i] = 0.f;
}

// ---------------- score = l2n(query) . mem^T, store column-major ----------------
__global__ __launch_bounds__(256) void k_score(const float* __restrict__ query,
                                               const float* __restrict__ mitems,
                                               float* __restrict__ ws) {
  __shared__ float meml[Mcl * Dch];
  int tid = threadIdx.x;
  for (int i = tid; i < Mcl * Dch; i += 256) meml[i] = mitems[i];
  __syncthreads();
  int n = blockIdx.x * 256 + tid;
  int b = n / HWq, hw = n % HWq;
  const float* qp = query + (size_t)b * CHW + hw;
  float ss = 0.f;
  float acc[Mcl];
#pragma unroll
  for (int m = 0; m < Mcl; ++m) acc[m] = 0.f;
  for (int d = 0; d < Dch; ++d) {
    float v = qp[(size_t)d * HWq];         // coalesced across n
    ss += v * v;
#pragma unroll
    for (int m = 0; m < Mcl; ++m) acc[m] += v * meml[m * Dch + d];
  }
  float inv = 1.0f / fmaxf(sqrtf(ss), 1e-12f);
  ws[WS_INV + n] = inv;
#pragma unroll
  for (int m = 0; m < Mcl; ++m) ws[WS_SCORE + (size_t)m * Ntot + n] = acc[m] * inv;
}

// ---------------- per-column (axis=0) max & sum-exp ----------------
__global__ __launch_bounds__(256) void k_colred(float* __restrict__ ws) {
  __shared__ float red[256];
  int m = blockIdx.x, tid = threadIdx.x;
  const float* col = ws + WS_SCORE + (size_t)m * Ntot;
  float mx = -1e30f;
  for (int i = tid; i < Ntot; i += 256) mx = fmaxf(mx, col[i]);
  red[tid] = mx; __syncthreads();
  for (int s = 128; s > 0; s >>= 1) { if (tid < s) red[tid] = fmaxf(red[tid], red[tid + s]); __syncthreads(); }
  mx = red[0]; __syncthreads();
  float sm = 0.f;
  for (int i = tid; i < Ntot; i += 256) sm += expf(col[i] - mx);
  red[tid] = sm; __syncthreads();
  for (int s = 128; s > 0; s >>= 1) { if (tid < s) red[tid] += red[tid + s]; __syncthreads(); }
  if (tid == 0) { ws[WS_CMAX + m] = mx; ws[WS_CSUM + m] = red[0]; }
}

// ---------------- row softmax (score_memory) + column softmax (score_query) ----------------
__global__ __launch_bounds__(256) void k_rowsoft(const float* __restrict__ ws,
                                                 float* __restrict__ out) {
  __shared__ float cmx[Mcl], csm[Mcl];
  int tid = threadIdx.x;
  if (tid < Mcl) { cmx[tid] = ws[WS_CMAX + tid]; csm[tid] = ws[WS_CSUM + tid]; }
  __syncthreads();
  int n = blockIdx.x * 256 + tid;
  float s[Mcl];
  float mx = -1e30f;
#pragma unroll
  for (int m = 0; m < Mcl; ++m) { s[m] = ws[WS_SCORE + (size_t)m * Ntot + n]; mx = fmaxf(mx, s[m]); }
  float sum = 0.f;
  float e[Mcl];
#pragma unroll
  for (int m = 0; m < Mcl; ++m) { e[m] = expf(s[m] - mx); sum += e[m]; }
  float inv = 1.f / sum;
#pragma unroll
  for (int m = 0; m < Mcl; ++m) {
    out[OUT_SM + (size_t)n * Mcl + m] = e[m] * inv;
    out[OUT_SQ + (size_t)n * Mcl + m] = expf(s[m] - cmx[m]) / csm[m];
  }
}

// ---------------- readloss: on-the-fly 8x bilinear upsample + CE ----------------
__global__ __launch_bounds__(256) void k_readloss(const float* __restrict__ ws,
                                                  const int* __restrict__ mask,
                                                  float* __restrict__ wsrl) {
  __shared__ float s0[Mcl * Wqq], s1[Mcl * Wqq];
  __shared__ float redn[256], redc[256];
  int blk = blockIdx.x;
  int jc = blk % 3; int i = (blk / 3) % Hm; int b = blk / (3 * Hm);
  float rr = (float)i * (95.0f / 767.0f);
  int r0 = (int)floorf(rr); r0 = min(max(r0, 0), 94);
  float wr = rr - (float)r0;
  int tid = threadIdx.x;
  for (int t = tid; t < Mcl * Wqq; t += 256) {
    int m = t / Wqq, c = t % Wqq;
    size_t base = WS_SCORE + (size_t)m * Ntot + (size_t)b * HWq;
    s0[t] = ws[base + (size_t)r0 * Wqq + c];
    s1[t] = ws[base + (size_t)(r0 + 1) * Wqq + c];
  }
  __syncthreads();
  int j = jc * 256 + tid;
  float cxr = (float)j * (95.0f / 767.0f);
  int c0 = (int)floorf(cxr); c0 = min(max(c0, 0), 94);
  float wc = cxr - (float)c0;
  float w00 = (1 - wr) * (1 - wc), w01 = (1 - wr) * wc, w10 = wr * (1 - wc), w11 = wr * wc;
  float lg[Mcl]; float mx = -1e30f;
#pragma unroll
  for (int m = 0; m < Mcl; ++m) {
    float v = w00 * s0[m * Wqq + c0] + w01 * s0[m * Wqq + c0 + 1]
            + w10 * s1[m * Wqq + c0] + w11 * s1[m * Wqq + c0 + 1];
    v *= TEMP_INV;
    lg[m] = v; mx = fmaxf(mx, v);
  }
  float se = 0.f;
#pragma unroll
  for (int m = 0; m < Mcl; ++m) se += expf(lg[m] - mx);
  float lse = mx + logf(se);
  int tgt = mask[(size_t)b * Hm * Hm + (size_t)i * Hm + j];
  bool valid = (tgt != 255);
  int tt = (tgt >= 0 && tgt < Mcl) ? tgt : 0;
  float nll = valid ? (lse - lg[tt]) : 0.f;
  redn[tid] = nll; redc[tid] = valid ? 1.f : 0.f;
  __syncthreads();
  for (int s = 128; s > 0; s >>= 1) {
    if (tid < s) { redn[tid] += redn[tid + s]; redc[tid] += redc[tid + s]; }
    __syncthreads();
  }
  if (tid == 0) { atomicAdd(&wsrl[WS_RL], redn[0]); atomicAdd(&wsrl[WS_RL + 1], redc[0]); }
}

// ---------------- P2 = mem @ Wm^T (tiny, row 19 zero pad) ----------------
__global__ __launch_bounds__(256) void k_p2(const float* __restrict__ mitems,
                                            const float* __restrict__ w_out,
                                            float* __restrict__ ws) {
  __shared__ float meml[Mcl * Dch];
  int tid = threadIdx.x;
  for (int i = tid; i < Mcl * Dch; i += 256) meml[i] = mitems[i];
  __syncthreads();
  int o = tid;
  float acc[Mcl];
#pragma unroll
  for (int m = 0; m < Mcl; ++m) acc[m] = 0.f;
  const float* wrow = w_out + (size_t)o * (2 * Dch) + Dch;
  for (int c = 0; c < Dch; ++c) {
    float wv = wrow[c];
#pragma unroll
    for (int m = 0; m < Mcl; ++m) acc[m] += wv * meml[m * Dch + c];
  }
#pragma unroll
  for (int m = 0; m < Mcl; ++m) ws[WS_P2 + m * Dch + o] = acc[m];
  ws[WS_P2 + Mcl * Dch + o] = 0.f;
}

// ---------------- w_write GEMM (WMMA f32 16x16x4) + residual/BN/ReLU/L2 + nominator ----------------
__global__ __launch_bounds__(256) void k_wwrite(const float* __restrict__ query,
                                                const int* __restrict__ mask,
                                                const float* __restrict__ w_write,
                                                const float* __restrict__ g, const float* __restrict__ bb,
                                                const float* __restrict__ mm, const float* __restrict__ vv,
                                                float* __restrict__ ws) {
  __shared__ float qtile[32 * QSTR];
  __shared__ float Bl[32 * BSTR];
  __shared__ float wql[32 * WSTR];
  __shared__ float noml[20 * Dch];
  __shared__ float tapw[32][4];
  __shared__ int   tapk[32][4];
  __shared__ float invn[32];
  __shared__ float denl[20];
  int tid = threadIdx.x;
  int wave = tid >> 5, lane = tid & 31, l16 = lane & 15, lhi = lane >> 4;
  int rowBase = blockIdx.x * 32;
  int b = rowBase / HWq, hwBase = rowBase % HWq;
  for (int i = tid; i < 20 * Dch; i += 256) noml[i] = 0.f;
  if (tid < 20) denl[tid] = 0.f;
  // stage raw query tile 32x256 (A matrix + residual), coalesced
  {
    int r = tid & 31, c0 = (tid >> 5) * 32;
    const float* qp = query + (size_t)b * CHW + hwBase + r;
    for (int c = c0; c < c0 + 32; ++c) qtile[r * QSTR + c] = qp[(size_t)c * HWq];
  }
  __syncthreads();
  v8f acc[4];
  {
    v8f z = {0.f, 0.f, 0.f, 0.f, 0.f, 0.f, 0.f, 0.f};
#pragma unroll
    for (int i = 0; i < 4; ++i) acc[i] = z;
  }
  int colBase = wave * 32;
  for (int kc = 0; kc < Dch; kc += 32) {
    {  // stage B chunk: Bl[k][o] = w_write[o][kc+k]
      const float* wr = w_write + (size_t)tid * Dch + kc;
#pragma unroll
      for (int k = 0; k < 32; ++k) Bl[k * BSTR + tid] = wr[k];
    }
    __syncthreads();
#pragma unroll
    for (int k0 = 0; k0 < 32; k0 += 4) {
      v2f bf[2];
#pragma unroll
      for (int cb = 0; cb < 2; ++cb) {
        int o = colBase + cb * 16 + l16; int kk = k0 + 2 * lhi;
        bf[cb].x = Bl[kk * BSTR + o]; bf[cb].y = Bl[(kk + 1) * BSTR + o];
      }
#pragma unroll
      for (int rb = 0; rb < 2; ++rb) {
        int r = rb * 16 + l16; int kk = kc + k0 + 2 * lhi;
        v2f af; af.x = qtile[r * QSTR + kk]; af.y = qtile[r * QSTR + kk + 1];
#pragma unroll
        for (int cb = 0; cb < 2; ++cb)
          acc[rb * 2 + cb] = __builtin_amdgcn_wmma_f32_16x16x4_f32(
              false, af, false, bf[cb], (short)0, acc[rb * 2 + cb], false, false);
      }
    }
    __syncthreads();
  }
  // epilogue: wq = relu(q + bn(pre)) -> LDS
#pragma unroll
  for (int rb = 0; rb < 2; ++rb)
#pragma unroll
    for (int cb = 0; cb < 2; ++cb) {
      int o = colBase + cb * 16 + l16;
      float s = g[o] * rsqrtf(vv[o] + BN_EPS);
      float t = bb[o] - mm[o] * s;
      v8f a = acc[rb * 2 + cb];
#pragma unroll
      for (int j = 0; j < 8; ++j) {
        int r = rb * 16 + j + 8 * lhi;
        float val = qtile[r * QSTR + o] + a[j] * s + t;
        wql[r * WSTR + o] = fmaxf(val, 0.f);
      }
    }
  __syncthreads();
  // row L2 norms + one-hot downsample taps (threads 0..31)
  if (tid < 32) {
    int r = tid;
    float ss = 0.f;
    for (int c = 0; c < Dch; ++c) { float v = wql[r * WSTR + c]; ss += v * v; }
    invn[r] = 1.0f / fmaxf(sqrtf(ss), 1e-12f);
    int hw = hwBase + r;
    int h = hw / Wqq, w = hw % Wqq;
    float rr = (float)h * (767.0f / 95.0f);
    int r0 = min(max((int)floorf(rr), 0), 766); float fr = rr - (float)r0;
    float cc = (float)w * (767.0f / 95.0f);
    int c0 = min(max((int)floorf(cc), 0), 766); float fc = cc - (float)c0;
    const int* mp = mask + (size_t)b * Hm * Hm;
    int kk[4]; float wt[4];
    kk[0] = mp[(size_t)r0 * Hm + c0];       wt[0] = (1 - fr) * (1 - fc);
    kk[1] = mp[(size_t)r0 * Hm + c0 + 1];   wt[1] = (1 - fr) * fc;
    kk[2] = mp[(size_t)(r0 + 1) * Hm + c0];     wt[2] = fr * (1 - fc);
    kk[3] = mp[(size_t)(r0 + 1) * Hm + c0 + 1]; wt[3] = fr * fc;
#pragma unroll
    for (int j = 0; j < 4; ++j) {
      int k = ((unsigned)kk[j] < 20u) ? kk[j] : 0;
      float wv = ((unsigned)kk[j] < 20u) ? wt[j] : 0.f;
      tapk[r][j] = k; tapw[r][j] = wv;
      atomicAdd(&denl[k], wv);
    }
  }
  __syncthreads();
  // nominator: nom[k][d] += w * wqn[r][d]  (thread owns channel d)
  {
    int d = tid;
    for (int r = 0; r < 32; ++r) {
      float v = wql[r * WSTR + d] * invn[r];
#pragma unroll
      for (int j = 0; j < 4; ++j)
        noml[tapk[r][j] * Dch + d] += tapw[r][j] * v;
    }
  }
  __syncthreads();
  for (int i = tid; i < 20 * Dch; i += 256) atomicAdd(&ws[WS_NOM + i], noml[i]);
  if (tid < 20) atomicAdd(&ws[WS_DEN + tid], denl[tid]);
}

// ---------------- w_out GEMM: qn@Wq^T + sm@P2 (rank-20 via WMMA), BN/ReLU, write updated_query ----------------
__global__ __launch_bounds__(256) void k_outgemm(const float* __restrict__ query,
                                                 const float* __restrict__ w_out,
                                                 const float* __restrict__ g, const float* __restrict__ bb,
                                                 const float* __restrict__ mm, const float* __restrict__ vv,
                                                 const float* __restrict__ ws,
                                                 float* out) {
  __shared__ float qtile[32 * QSTR];
  __shared__ float Bl[32 * BSTR];
  __shared__ float smt[32 * SMSTR];
  __shared__ float p2l[20 * PSTR];
  int tid = threadIdx.x;
  int wave = tid >> 5, lane = tid & 31, l16 = lane & 15, lhi = lane >> 4;
  int rowBase = blockIdx.x * 32;
  int b = rowBase / HWq, hwBase = rowBase % HWq;
  // stage qn = query * inv_norm
  {
    int r = tid & 31, c0 = (tid >> 5) * 32;
    float inv = ws[WS_INV + rowBase + r];
    const float* qp = query + (size_t)b * CHW + hwBase + r;
    for (int c = c0; c < c0 + 32; ++c) qtile[r * QSTR + c] = qp[(size_t)c * HWq] * inv;
  }
  for (int i = tid; i < 32 * SMSTR; i += 256) smt[i] = 0.f;
  __syncthreads();
  for (int i = tid; i < 32 * Mcl; i += 256) {
    int r = i / Mcl, m = i % Mcl;
    smt[r * SMSTR + m] = out[OUT_SM + (size_t)(rowBase + r) * Mcl + m];
  }
  for (int i = tid; i < 20 * Dch; i += 256) {
    int m = i / Dch, o = i % Dch;
    p2l[m * PSTR + o] = ws[WS_P2 + i];
  }
  __syncthreads();
  v8f acc[4];
  {
    v8f z = {0.f, 0.f, 0.f, 0.f, 0.f, 0.f, 0.f, 0.f};
#pragma unroll
    for (int i = 0; i < 4; ++i) acc[i] = z;
  }
  int colBase = wave * 32;
  for (int kc = 0; kc < Dch; kc += 32) {
    {  // Bl[k][o] = w_out[o][kc+k]  (first 256 input channels)
      const float* wr = w_out + (size_t)tid * (2 * Dch) + kc;
#pragma unroll
      for (int k = 0; k < 32; ++k) Bl[k * BSTR + tid] = wr[k];
    }
    __syncthreads();
#pragma unroll
    for (int k0 = 0; k0 < 32; k0 += 4) {
      v2f bf[2];
#pragma unroll
      for (int cb = 0; cb < 2; ++cb) {
        int o = colBase + cb * 16 + l16; int kk = k0 + 2 * lhi;
        bf[cb].x = Bl[kk * BSTR + o]; bf[cb].y = Bl[(kk + 1) * BSTR + o];
      }
#pragma unroll
      for (int rb = 0; rb < 2; ++rb) {
        int r = rb * 16 + l16; int kk = kc + k0 + 2 * lhi;
        v2f af; af.x = qtile[r * QSTR + kk]; af.y = qtile[r * QSTR + kk + 1];
#pragma unroll
        for (int cb = 0; cb < 2; ++cb)
          acc[rb * 2 + cb] = __builtin_amdgcn_wmma_f32_16x16x4_f32(
              false, af, false, bf[cb], (short)0, acc[rb * 2 + cb], false, false);
      }
    }
    __syncthreads();
  }
  // rank-20 correction: + sm(32x20) @ P2(20x256) via 5 more WMMA k-steps
#pragma unroll
  for (int k0 = 0; k0 < 20; k0 += 4) {
    v2f bf[2];
#pragma unroll
    for (int cb = 0; cb < 2; ++cb) {
      int o = colBase + cb * 16 + l16; int kk = k0 + 2 * lhi;
      bf[cb].x = p2l[kk * PSTR + o]; bf[cb].y = p2l[(kk + 1) * PSTR + o];
    }
#pragma unroll
    for (int rb = 0; rb < 2; ++rb) {
      int r = rb * 16 + l16; int kk = k0 + 2 * lhi;
      v2f af; af.x = smt[r * SMSTR + kk]; af.y = smt[r * SMSTR + kk + 1];
#pragma unroll
      for (int cb = 0; cb < 2; ++cb)
        acc[rb * 2 + cb] = __builtin_amdgcn_wmma_f32_16x16x4_f32(
            false, af, false, bf[cb], (short)0, acc[rb * 2 + cb], false, false);
    }
  }
  __syncthreads();
  // epilogue: relu(bn(pre)) -> LDS (reuse Bl) -> coalesced BCHW store
#pragma unroll
  for (int rb = 0; rb < 2; ++rb)
#pragma unroll
    for (int cb = 0; cb < 2; ++cb) {
      int o = colBase + cb * 16 + l16;
      float s = g[o] * rsqrtf(vv[o] + BN_EPS);
      float t = bb[o] - mm[o] * s;
      v8f a = acc[rb * 2 + cb];
#pragma unroll
      for (int j = 0; j < 8; ++j) {
        int r = rb * 16 + j + 8 * lhi;
        Bl[r * BSTR + o] = fmaxf(a[j] * s + t, 0.f);
      }
    }
  __syncthreads();
  for (int i = tid; i < 32 * Dch; i += 256) {
    int o = i >> 5, r = i & 31;
    out[OUT_UQ + (size_t)b * CHW + (size_t)o * HWq + hwBase + r] = Bl[r * BSTR + o];
  }
}

// ---------------- memory update + div_loss + cls_loss + finalize readloss ----------------
__global__ __launch_bounds__(256) void k_final(const float* __restrict__ mitems,
                                               const float* __restrict__ cw,
                                               const float* __restrict__ cbv,
                                               float* __restrict__ ws,
                                               float* __restrict__ out) {
  __shared__ float um[Mcl * Dch];
  __shared__ float lg[Mcl * Mcl];
  __shared__ float rn[Mcl];
  __shared__ float accDiv, accCls;
  int tid = threadIdx.x;
  if (tid == 0) { accDiv = 0.f; accCls = 0.f; }
  for (int i = tid; i < Mcl * Dch; i += 256) {
    int k = i / Dch;
    float den = ws[WS_DEN + k];
    float nm = 0.9f * mitems[i] + 0.1f * ws[WS_NOM + i] / fmaxf(den, 1e-12f);
    um[i] = (den != 0.f) ? nm : mitems[i];
  }
  __syncthreads();
  if (tid < Mcl) {
    float ss = 0.f;
    for (int d = 0; d < Dch; ++d) { float v = um[tid * Dch + d]; ss += v * v; }
    rn[tid] = 1.f / fmaxf(sqrtf(ss), 1e-12f);
  }
  __syncthreads();
  for (int i = tid; i < Mcl * Dch; i += 256) um[i] *= rn[i / Dch];
  __syncthreads();
  for (int i = tid; i < Mcl * Mcl; i += 256) {
    int a = i / Mcl, c = i % Mcl;
    float dcos = 0.f, dcls = 0.f;
    for (int d = 0; d < Dch; ++d) {
      dcos += um[a * Dch + d] * um[c * Dch + d];
      dcls += um[a * Dch + d] * cw[c * Dch + d];
    }
    lg[i] = dcls + cbv[c];
    if (a != c) atomicAdd(&accDiv, fmaxf(dcos, 0.f));
  }
  __syncthreads();
  if (tid < Mcl) {
    float mx = -1e30f;
    for (int c = 0; c < Mcl; ++c) mx = fmaxf(mx, lg[tid * Mcl + c]);
    float se = 0.f;
    for (int c = 0; c < Mcl; ++c) se += expf(lg[tid * Mcl + c] - mx);
    float nll = mx + logf(se) - lg[tid * Mcl + tid];
    atomicAdd(&accCls, nll);
  }
  __syncthreads();
  if (tid == 0) {
    out[OUT_LS + 0] = ws[WS_RL] / fmaxf(ws[WS_RL + 1], 1.0f);
    out[OUT_LS + 1] = accDiv / (float)(Mcl * (Mcl - 1));
    out[OUT_LS + 2] = accCls / (float)Mcl;
  }
}

extern "C" void kernel_launch(void* const* d_in, const int* in_sizes, int n_in,
                              void* d_out, int out_size, void* d_ws, size_t ws_size,
                              hipStream_t stream) {
  (void)in_sizes; (void)n_in; (void)out_size; (void)ws_size;
  const float* query   = (const float*)d_in[0];
  const int*   mask    = (const int*)d_in[1];
  const float* mitems  = (const float*)d_in[2];
  const float* w_write = (const float*)d_in[3];
  const float* wbn_g   = (const float*)d_in[4];
  const float* wbn_b   = (const float*)d_in[5];
  const float* wbn_m   = (const float*)d_in[6];
  const float* wbn_v   = (const float*)d_in[7];
  const float* w_out   = (const float*)d_in[8];
  const float* obn_g   = (const float*)d_in[9];
  const float* obn_b   = (const float*)d_in[10];
  const float* obn_m   = (const float*)d_in[11];
  const float* obn_v   = (const float*)d_in[12];
  const float* cw      = (const float*)d_in[13];
  const float* cbv     = (const float*)d_in[14];
  float* out = (float*)d_out;
  float* ws  = (float*)d_ws;

  k_init<<<21, 256, 0, stream>>>(ws);
  k_score<<<Ntot / 256, 256, 0, stream>>>(query, mitems, ws);
  k_colred<<<Mcl, 256, 0, stream>>>(ws);
  k_rowsoft<<<Ntot / 256, 256, 0, stream>>>(ws, out);
  k_readloss<<<Bsz * Hm * 3, 256, 0, stream>>>(ws, mask, ws);
  k_p2<<<1, 256, 0, stream>>>(mitems, w_out, ws);
  k_wwrite<<<Ntot / 32, 256, 0, stream>>>(query, mask, w_write, wbn_g, wbn_b, wbn_m, wbn_v, ws);
  k_outgemm<<<Ntot / 32, 256, 0, stream>>>(query, w_out, obn_g, obn_b, obn_m, obn_v, ws, out);
  k_final<<<1, 256, 0, stream>>>(mitems, cw, cbv, ws, out);
}